// PhaseAwareBinaryAttention_50122268344533
// MI455X (gfx1250) — compile-verified
//
#include <hip/hip_runtime.h>
#include <math.h>

typedef float v2f __attribute__((ext_vector_type(2)));
typedef float v8f __attribute__((ext_vector_type(8)));

#define NT 2048

// ---------------- Pass 1: per-block partial sums of dist and dist^2 ----------
// grid = 4 batches * 32 blocks, 256 threads. Each block: 64 s-rows x 2048 t.
__global__ __launch_bounds__(256) void paba_pass1(const float* __restrict__ r,
                                                  double* __restrict__ part) {
  __shared__ float4 l4[NT];        // whole batch embedding, 32 KB
  __shared__ double red1[256];
  __shared__ double red2[256];
  const int b  = blockIdx.x >> 5;
  const int s0 = (blockIdx.x & 31) * 64;
  const float4* rb = (const float4*)(r + (size_t)b * NT * 4);
  for (int i = threadIdx.x; i < NT; i += 256) l4[i] = rb[i];
  __syncthreads();

  double a1 = 0.0, a2 = 0.0;
  for (int idx = threadIdx.x; idx < 64 * NT; idx += 256) {
    const int sl = idx >> 11;
    const int t  = idx & (NT - 1);
    const float4 rs = l4[s0 + sl];
    const float4 rt = l4[t];
    const float o0 = 1.f - rt.x, o1 = 1.f - rt.y, o2 = 1.f - rt.z, o3 = 1.f - rt.w;
    const float d = (rs.x * rs.x) * (o0 * o0) + (rs.y * rs.y) * (o1 * o1) +
                    (rs.z * rs.z) * (o2 * o2) + (rs.w * rs.w) * (o3 * o3);
    a1 += (double)d;
    a2 += (double)d * (double)d;
  }
  red1[threadIdx.x] = a1;
  red2[threadIdx.x] = a2;
  __syncthreads();
  for (int off = 128; off > 0; off >>= 1) {
    if (threadIdx.x < off) {
      red1[threadIdx.x] += red1[threadIdx.x + off];
      red2[threadIdx.x] += red2[threadIdx.x + off];
    }
    __syncthreads();
  }
  if (threadIdx.x == 0) {
    part[blockIdx.x * 2 + 0] = red1[0];
    part[blockIdx.x * 2 + 1] = red2[0];
  }
}

// ---------------- Pass 1b: threshold[b] = mean + 1.25 * clip(std, 1e-6) -----
__global__ __launch_bounds__(128) void paba_pass1b(const double* __restrict__ part,
                                                   float* __restrict__ thr) {
  const int b    = threadIdx.x >> 5;   // 4 batches, one wave each
  const int lane = threadIdx.x & 31;
  double s1 = part[(b * 32 + lane) * 2 + 0];
  double s2 = part[(b * 32 + lane) * 2 + 1];
  for (int off = 16; off > 0; off >>= 1) {
    s1 += __shfl_xor(s1, off, 32);
    s2 += __shfl_xor(s2, off, 32);
  }
  if (lane == 0) {
    const double N = (double)NT * (double)NT;   // 4194304
    const double mean = s1 / N;
    double var = (s2 - s1 * s1 / N) / (N - 1.0);
    if (var < 0.0) var = 0.0;
    double sd = sqrt(var);
    if (sd < 1e-6) sd = 1e-6;
    thr[b] = (float)(mean + 1.25 * sd);
  }
}

// ---------------- Pass 2: WMMA dist tiles -> attention -> row-normalize -----
// blockDim = 64 (2 waves). Each wave owns a 16(s) x 2048(t) strip.
// grid = 4 batches * 64 blocks.
__global__ __launch_bounds__(64) void paba_pass2(const float* __restrict__ r,
                                                 const float* __restrict__ thr,
                                                 float* __restrict__ out) {
  __shared__ float4 l4[NT];       // batch embeddings, 32 KB
  __shared__ float  linv[2][16];  // per-row 1/(rowsum+eps)
  const int b      = blockIdx.x >> 6;
  const int sblock = (blockIdx.x & 63) * 32;
  const int wave   = threadIdx.x >> 5;
  const int lane   = threadIdx.x & 31;
  const int hi     = lane >> 4;       // lane half selects K={2,3} vs K={0,1}
  const int m      = lane & 15;

  const float4* rb = (const float4*)(r + (size_t)b * NT * 4);
  for (int i = threadIdx.x; i < NT; i += 64) l4[i] = rb[i];
  __syncthreads();

  const float threshold = thr[b];
  const int sbase = sblock + wave * 16;

  // A operand: asq[m][k]  (16x4 f32 A layout: lanes 0-15 K0/K1, lanes 16-31 K2/K3)
  const float4 rsm = l4[sbase + m];
  v2f a;
  a.x = hi ? rsm.z * rsm.z : rsm.x * rsm.x;
  a.y = hi ? rsm.w * rsm.w : rsm.y * rsm.y;

  float* outB = out + (size_t)b * NT * NT;
  float rs0v[8], rs1v[8], rowsum[8];
  float* prow[8];
#pragma unroll
  for (int v = 0; v < 8; ++v) {       // C-layout rows: s = sbase + v + 8*hi
    const int s = sbase + v + 8 * hi;
    const float4 q = l4[s];
    rs0v[v] = q.x;
    rs1v[v] = q.y;
    rowsum[v] = 0.f;
    prow[v] = outB + (size_t)s * NT;
  }

#pragma unroll 2
  for (int tt = 0; tt < NT; tt += 16) {
    const int t = tt + m;
    const float4 rt4 = l4[t];
    const float o0 = 1.f - rt4.x, o1 = 1.f - rt4.y;
    const float o2 = 1.f - rt4.z, o3 = 1.f - rt4.w;
    // B operand: bsq[k][n]  (4x16 f32 B layout mirrors A's lane-half K split)
    v2f bv;
    bv.x = hi ? o2 * o2 : o0 * o0;
    bv.y = hi ? o3 * o3 : o1 * o1;
    v8f c = {};
    // dist tile = A(rs^2) x B((1-rt)^2): one 16x16x4 f32 WMMA
    c = __builtin_amdgcn_wmma_f32_16x16x4_f32(false, a, false, bv,
                                              (short)0, c, false, false);
#pragma unroll
    for (int v = 0; v < 8; ++v) {
      const float d  = c[v];
      const float d0 = rs0v[v] * o0;
      const float d1 = rs1v[v] * o1;
      const float r2 = d0 * d0 + d1 * d1;
      // branchless cos(atan2(d1,d0)): raw v_rsq_f32, cndmask for the (0,0) case
      const float csr = d0 * __builtin_amdgcn_rsqf(r2);
      const float cs  = (r2 > 0.f) ? csr : 1.f;
      const float bias = 0.5f * (1.f + cs) * __expf(-d);
      const float att  = (d <= threshold) ? bias : 0.f;
      rowsum[v] += att;
      prow[v][t] = att;            // unnormalized, stays hot in L2
    }
  }

  // reduce row sums across each 16-lane half (same s set per half)
#pragma unroll
  for (int v = 0; v < 8; ++v) {
    float s = rowsum[v];
    s += __shfl_xor(s, 8, 16);
    s += __shfl_xor(s, 4, 16);
    s += __shfl_xor(s, 2, 16);
    s += __shfl_xor(s, 1, 16);
    rowsum[v] = 1.f / (s + 1e-8f);
  }
  if (m == 0) {
#pragma unroll
    for (int v = 0; v < 8; ++v) linv[wave][8 * hi + v] = rowsum[v];
  }
  __threadfence_block();   // make this wave's global stores WG-visible
  __syncthreads();

  // in-place rescale of own 16x2048 strip with wide float4 accesses (L2 hits)
  float4* o4 = (float4*)(outB + (size_t)sbase * NT);
  for (int i = lane; i < 16 * (NT / 4); i += 32) {
    const int sl = i >> 9;               // row within strip
    const float sc = linv[wave][sl];
    float4 x = o4[i];
    x.x *= sc; x.y *= sc; x.z *= sc; x.w *= sc;
    o4[i] = x;
  }
}

extern "C" void kernel_launch(void* const* d_in, const int* in_sizes, int n_in,
                              void* d_out, int out_size, void* d_ws, size_t ws_size,
                              hipStream_t stream) {
  (void)in_sizes; (void)n_in; (void)out_size; (void)ws_size;
  const float* r = (const float*)d_in[0];
  float* out = (float*)d_out;
  double* part = (double*)d_ws;                               // 128 blocks * 2 doubles
  float*  thr  = (float*)((char*)d_ws + 128 * 2 * sizeof(double));

  paba_pass1 <<<128, 256, 0, stream>>>(r, part);
  paba_pass1b<<<1,   128, 0, stream>>>(part, thr);
  paba_pass2 <<<4 * 64, 64, 0, stream>>>(r, thr, out);
}